// SATLayer_regular_43731357008210
// MI455X (gfx1250) — compile-verified
//
#include <hip/hip_runtime.h>
#include <hip/hip_bf16.h>

// ---------------------------------------------------------------------------
// SAT layer for MI455X (gfx1250, wave32).
// GEMMs use V_WMMA_F32_16X16X4_F32 (full fp32 fidelity; GEMM is not the
// bottleneck -- the 1.6M-edge gather/scatter phase is, and it lives in L2).
// ---------------------------------------------------------------------------

typedef __attribute__((ext_vector_type(2))) float v2f;
typedef __attribute__((ext_vector_type(8))) float v8f;

#define DFEAT 128
#define LN_EPS 1e-5f

__device__ __forceinline__ float sigmoidf_(float x) {
  return 1.0f / (1.0f + __expf(-x));
}

__device__ __forceinline__ float leaky_(float x) {
  return x >= 0.0f ? x : 0.2f * x;
}

// ---------------------------------------------------------------------------
// Kernel 1: xi = leaky(x0 @ W1 + b1), xj = leaky(x0 @ W2 + b2)
// One block = 16 output rows; 8 wave32s each own one 16-wide column tile.
// A fragments (x0 rows) are shared by the W1 and W2 accumulators.
// WMMA f32 16x16x4 layout (ISA 7.12.2):
//   A lane L<16:  a = {A[L][k0+0], A[L][k0+1]};  lane L+16: {A[L][k0+2], A[L][k0+3]}
//   B lane L<16:  b = {B[k0+0][L], B[k0+1][L]};  lane L+16: {B[k0+2][L], B[k0+3][L]}
//   C/D vgpr v:   row = v + 8*(lane>>4), col = lane&15
// ---------------------------------------------------------------------------
__global__ __launch_bounds__(256) void node_transform_kernel(
    const float* __restrict__ x0,
    const float* __restrict__ W1, const float* __restrict__ b1,
    const float* __restrict__ W2, const float* __restrict__ b2,
    float* __restrict__ xi, float* __restrict__ xj, int N)
{
  const int m0   = blockIdx.x * 16;
  const int wave = threadIdx.x >> 5;
  const int lane = threadIdx.x & 31;
  const int n0   = wave * 16;
  const int r16  = lane & 15;
  const int kh   = lane >> 4;        // which K half-pair this lane holds

  int arow = m0 + r16;
  if (arow >= N) arow = N - 1;       // clamp loads; stores are guarded
  const float* __restrict__ Ar = x0 + (size_t)arow * DFEAT;

  v8f acc1 = {};
  v8f acc2 = {};

#pragma unroll
  for (int k0 = 0; k0 < DFEAT; k0 += 4) {
    const int ka = k0 + 2 * kh;
    v2f a;
    a.x = Ar[ka];
    a.y = Ar[ka + 1];
    v2f bA, bB;
    bA.x = W1[(size_t)ka * DFEAT + n0 + r16];
    bA.y = W1[(size_t)(ka + 1) * DFEAT + n0 + r16];
    bB.x = W2[(size_t)ka * DFEAT + n0 + r16];
    bB.y = W2[(size_t)(ka + 1) * DFEAT + n0 + r16];
    acc1 = __builtin_amdgcn_wmma_f32_16x16x4_f32(false, a, false, bA,
                                                 (short)0, acc1, false, false);
    acc2 = __builtin_amdgcn_wmma_f32_16x16x4_f32(false, a, false, bB,
                                                 (short)0, acc2, false, false);
  }

  const int col = n0 + r16;
  const float bb1 = b1[col];
  const float bb2 = b2[col];
#pragma unroll
  for (int v = 0; v < 8; ++v) {
    const int row = m0 + v + 8 * kh;
    if (row < N) {
      xi[(size_t)row * DFEAT + col] = leaky_(acc1[v] + bb1);
      xj[(size_t)row * DFEAT + col] = leaky_(acc2[v] + bb2);
    }
  }
}

// ---------------------------------------------------------------------------
// Kernel 2: ai = xi @ a1_w + a1_b ; aj = xj @ a2_w + a2_b  (one wave32/node)
// ---------------------------------------------------------------------------
__global__ __launch_bounds__(256) void node_scores_kernel(
    const float* __restrict__ xi, const float* __restrict__ xj,
    const float* __restrict__ a1_w, const float* __restrict__ a1_b,
    const float* __restrict__ a2_w, const float* __restrict__ a2_b,
    float* __restrict__ ai, float* __restrict__ aj, int N)
{
  const int node = blockIdx.x * 8 + (threadIdx.x >> 5);
  const int lane = threadIdx.x & 31;
  if (node >= N) return;

  float s1 = 0.0f, s2 = 0.0f;
#pragma unroll
  for (int k = lane; k < DFEAT; k += 32) {
    s1 += xi[(size_t)node * DFEAT + k] * a1_w[k];
    s2 += xj[(size_t)node * DFEAT + k] * a2_w[k];
  }
#pragma unroll
  for (int off = 16; off >= 1; off >>= 1) {
    s1 += __shfl_down(s1, off, 32);
    s2 += __shfl_down(s2, off, 32);
  }
  if (lane == 0) {
    ai[node] = s1 + a1_b[0];
    aj[node] = s2 + a2_b[0];
  }
}

// ---------------------------------------------------------------------------
// Kernel 3: zero the aggregation buffer
// ---------------------------------------------------------------------------
__global__ void zero_kernel(float* __restrict__ p, size_t n) {
  size_t i = (size_t)blockIdx.x * blockDim.x + threadIdx.x;
  if (i < n) p[i] = 0.0f;
}

// ---------------------------------------------------------------------------
// Kernel 4: edge phase. One wave32 per edge; lane holds a float4 of xj[col].
// xj (51 MB) and agg (51 MB) both fit in the 192 MB L2, so the random gather
// and the global_atomic_add_f32 scatter stay on-chip.
// ---------------------------------------------------------------------------
__global__ __launch_bounds__(256) void edge_kernel(
    const int* __restrict__ erow, const int* __restrict__ ecol,
    const float* __restrict__ ai, const float* __restrict__ aj,
    const float* __restrict__ xj, float* __restrict__ agg, int E)
{
  const int e = blockIdx.x * 8 + (threadIdx.x >> 5);
  if (e >= E) return;
  const int lane = threadIdx.x & 31;

  const int r = erow[e];
  const int c = ecol[e];
  const float att = sigmoidf_(ai[r] + aj[c]);   // broadcast load, all lanes

  const float4 v = ((const float4*)(xj + (size_t)c * DFEAT))[lane];
  float* dst = agg + (size_t)r * DFEAT + lane * 4;
  atomicAdd(dst + 0, att * v.x);
  atomicAdd(dst + 1, att * v.y);
  atomicAdd(dst + 2, att * v.z);
  atomicAdd(dst + 3, att * v.w);
}

// ---------------------------------------------------------------------------
// Kernel 5: out = sigmoid(agg) @ Wo + bo + xi ; h = out + x0 ; LayerNorm(h)
// Sigmoid fused into the A-fragment loads. Row tile staged in LDS for LN.
// LN: 16 threads per row, 8 elems each, __shfl_xor butterfly (width 16,
// wave32-safe: groups of 16 lanes never cross the wave).
// ---------------------------------------------------------------------------
__global__ __launch_bounds__(256) void output_ln_kernel(
    const float* __restrict__ agg, const float* __restrict__ Wo,
    const float* __restrict__ bo,  const float* __restrict__ xi,
    const float* __restrict__ x0,  const float* __restrict__ gamma,
    const float* __restrict__ beta, float* __restrict__ out, int N)
{
  __shared__ float hbuf[16 * DFEAT];

  const int m0   = blockIdx.x * 16;
  const int wave = threadIdx.x >> 5;
  const int lane = threadIdx.x & 31;
  const int n0   = wave * 16;
  const int r16  = lane & 15;
  const int kh   = lane >> 4;

  int arow = m0 + r16;
  if (arow >= N) arow = N - 1;
  const float* __restrict__ Ar = agg + (size_t)arow * DFEAT;

  v8f acc = {};
#pragma unroll
  for (int k0 = 0; k0 < DFEAT; k0 += 4) {
    const int ka = k0 + 2 * kh;
    v2f a;
    a.x = sigmoidf_(Ar[ka]);
    a.y = sigmoidf_(Ar[ka + 1]);
    v2f b;
    b.x = Wo[(size_t)ka * DFEAT + n0 + r16];
    b.y = Wo[(size_t)(ka + 1) * DFEAT + n0 + r16];
    acc = __builtin_amdgcn_wmma_f32_16x16x4_f32(false, a, false, b,
                                                (short)0, acc, false, false);
  }

  const int col = n0 + r16;
  const float bb = bo[col];
#pragma unroll
  for (int v = 0; v < 8; ++v) {
    const int rloc = v + 8 * kh;
    const int row  = m0 + rloc;
    float h = acc[v] + bb;
    if (row < N)
      h += xi[(size_t)row * DFEAT + col] + x0[(size_t)row * DFEAT + col];
    hbuf[rloc * DFEAT + col] = h;
  }
  __syncthreads();

  // LayerNorm: thread t -> row t>>4, segment t&15 (8 contiguous elements).
  const int row = threadIdx.x >> 4;
  const int seg = threadIdx.x & 15;
  const float* __restrict__ hr = hbuf + row * DFEAT;

  float s = 0.0f;
#pragma unroll
  for (int j = 0; j < 8; ++j) s += hr[seg * 8 + j];
#pragma unroll
  for (int m = 8; m >= 1; m >>= 1) s += __shfl_xor(s, m, 16);
  const float mu = s * (1.0f / DFEAT);

  float vs = 0.0f;
#pragma unroll
  for (int j = 0; j < 8; ++j) {
    const float d = hr[seg * 8 + j] - mu;
    vs += d * d;
  }
#pragma unroll
  for (int m = 8; m >= 1; m >>= 1) vs += __shfl_xor(vs, m, 16);
  const float inv = rsqrtf(vs * (1.0f / DFEAT) + LN_EPS);

  const int grow = m0 + row;
  if (grow < N) {
#pragma unroll
    for (int j = 0; j < 8; ++j) {
      const int c = seg * 8 + j;
      out[(size_t)grow * DFEAT + c] = (hr[c] - mu) * inv * gamma[c] + beta[c];
    }
  }
}

// ---------------------------------------------------------------------------
// Launcher
// Inputs: 0 x0, 1 W1, 2 b1, 3 W2, 4 b2, 5 a1_w, 6 a1_b, 7 a2_w, 8 a2_b,
//         9 Wo, 10 bo, 11 ln_gamma, 12 ln_beta, 13 edge_rows, 14 edge_cols
// ---------------------------------------------------------------------------
extern "C" void kernel_launch(void* const* d_in, const int* in_sizes, int n_in,
                              void* d_out, int out_size, void* d_ws, size_t ws_size,
                              hipStream_t stream) {
  const float* x0   = (const float*)d_in[0];
  const float* W1   = (const float*)d_in[1];
  const float* b1   = (const float*)d_in[2];
  const float* W2   = (const float*)d_in[3];
  const float* b2   = (const float*)d_in[4];
  const float* a1w  = (const float*)d_in[5];
  const float* a1b  = (const float*)d_in[6];
  const float* a2w  = (const float*)d_in[7];
  const float* a2b  = (const float*)d_in[8];
  const float* Wo   = (const float*)d_in[9];
  const float* bo   = (const float*)d_in[10];
  const float* gam  = (const float*)d_in[11];
  const float* bet  = (const float*)d_in[12];
  const int*   erow = (const int*)d_in[13];
  const int*   ecol = (const int*)d_in[14];

  const int N = in_sizes[0] / DFEAT;
  const int E = in_sizes[13];
  float* outp = (float*)d_out;

  // Workspace layout (fp32): xi, xj, agg [N*D each], ai, aj [N each]
  float* xi  = (float*)d_ws;
  float* xj  = xi  + (size_t)N * DFEAT;
  float* agg = xj  + (size_t)N * DFEAT;
  float* ai  = agg + (size_t)N * DFEAT;
  float* aj  = ai  + N;

  const int mtiles = (N + 15) / 16;

  node_transform_kernel<<<mtiles, 256, 0, stream>>>(x0, W1, b1, W2, b2, xi, xj, N);
  node_scores_kernel<<<(N + 7) / 8, 256, 0, stream>>>(xi, xj, a1w, a1b, a2w, a2b, ai, aj, N);
  const size_t aggN = (size_t)N * DFEAT;
  zero_kernel<<<(unsigned)((aggN + 255) / 256), 256, 0, stream>>>(agg, aggN);
  edge_kernel<<<(E + 7) / 8, 256, 0, stream>>>(erow, ecol, ai, aj, xj, agg, E);
  output_ln_kernel<<<mtiles, 256, 0, stream>>>(agg, Wo, bo, xi, x0, gam, bet, outp, N);
}